// CrossPatientRetrieval_10333691314233
// MI455X (gfx1250) — compile-verified
//
#include <hip/hip_runtime.h>
#include <hip/hip_bf16.h>
#include <math.h>

typedef __attribute__((ext_vector_type(2))) float v2f;
typedef __attribute__((ext_vector_type(4))) float v4f;
typedef __attribute__((ext_vector_type(8))) float v8f;
typedef int v4i_gcc __attribute__((vector_size(16)));   // matches builtin param type

#define B_Q    64
#define C_DIM  256
#define N_BANK 20000
#define NT     64
#define TOPK   8
#define LDS_STRIDE 260   // 256 + 4 pad: A-frag ds_load_b64 hits 64 distinct banks

__device__ __forceinline__ float waveReduceSum(float v) {
  #pragma unroll
  for (int off = 16; off > 0; off >>= 1) v += __shfl_xor(v, off, 32);
  return v;
}

// ---------------- Kernel A: q / max(||q||, 1e-12) ----------------
__global__ void cpr_normalize_query(const float* __restrict__ q, float* __restrict__ qn) {
  const int b = blockIdx.x, t = threadIdx.x;
  float v = q[b * C_DIM + t];
  __shared__ float red[C_DIM];
  red[t] = v * v;
  __syncthreads();
  for (int s = 128; s > 0; s >>= 1) {
    if (t < s) red[t] += red[t + s];
    __syncthreads();
  }
  const float inv = 1.0f / fmaxf(sqrtf(red[0]), 1e-12f);
  qn[b * C_DIM + t] = v * inv;
}

// ---------------- Kernel B: 1 / max(||s_n||, 1e-12) (one wave32 per row) ----------------
__global__ void cpr_summary_invnorm(const float* __restrict__ s, float* __restrict__ sinv) {
  const int wave = threadIdx.x >> 5, lane = threadIdx.x & 31;
  const int row = blockIdx.x * 8 + wave;
  if (row >= N_BANK) return;
  const float* p = s + (size_t)row * C_DIM;
  float acc = 0.0f;
  #pragma unroll
  for (int q = 0; q < 8; q++) { float x = p[q * 32 + lane]; acc += x * x; }
  acc = waveReduceSum(acc);
  if (lane == 0) sinv[row] = 1.0f / fmaxf(sqrtf(acc), 1e-12f);
}

// ---------------- Kernel C: sims = qn @ s^T, post-scaled by sinv ----------------
// One wave per block; block computes 64 (queries) x 16 (bank cols) via fp32 WMMA 16x16x4.
// Bank row data streamed exactly once from HBM (20.5 MB ~ 1 us at 23.3 TB/s).
__global__ void cpr_sims_wmma(const float* __restrict__ qn, const float* __restrict__ s,
                              const float* __restrict__ sinv, float* __restrict__ sims) {
  const int lane  = threadIdx.x;
  const int n0    = blockIdx.x * 16;
  const int lr    = lane & 15;
  const int khalf = (lane >> 4) << 1;   // 0 or 2 per ISA fragment layout
  const int col   = n0 + lr;
  const float* srow = s + (size_t)col * C_DIM;

  v8f acc0 = {}, acc1 = {}, acc2 = {}, acc3 = {};
  for (int k0 = 0; k0 < C_DIM; k0 += 4) {
    const int kk = k0 + khalf;
    v2f bf; bf.x = srow[kk]; bf.y = srow[kk + 1];
    v2f a0, a1, a2, a3;
    a0.x = qn[(0 * 16 + lr) * C_DIM + kk]; a0.y = qn[(0 * 16 + lr) * C_DIM + kk + 1];
    a1.x = qn[(1 * 16 + lr) * C_DIM + kk]; a1.y = qn[(1 * 16 + lr) * C_DIM + kk + 1];
    a2.x = qn[(2 * 16 + lr) * C_DIM + kk]; a2.y = qn[(2 * 16 + lr) * C_DIM + kk + 1];
    a3.x = qn[(3 * 16 + lr) * C_DIM + kk]; a3.y = qn[(3 * 16 + lr) * C_DIM + kk + 1];
    acc0 = __builtin_amdgcn_wmma_f32_16x16x4_f32(false, a0, false, bf, (short)0, acc0, false, false);
    acc1 = __builtin_amdgcn_wmma_f32_16x16x4_f32(false, a1, false, bf, (short)0, acc1, false, false);
    acc2 = __builtin_amdgcn_wmma_f32_16x16x4_f32(false, a2, false, bf, (short)0, acc2, false, false);
    acc3 = __builtin_amdgcn_wmma_f32_16x16x4_f32(false, a3, false, bf, (short)0, acc3, false, false);
  }
  const float sv = sinv[col];
  const int rbase = (lane >> 4) << 3;   // C/D rows: r + 8*(lane>=16)
  #pragma unroll
  for (int r = 0; r < 8; r++) {
    sims[(size_t)(0 * 16 + rbase + r) * N_BANK + col] = acc0[r] * sv;
    sims[(size_t)(1 * 16 + rbase + r) * N_BANK + col] = acc1[r] * sv;
    sims[(size_t)(2 * 16 + rbase + r) * N_BANK + col] = acc2[r] * sv;
    sims[(size_t)(3 * 16 + rbase + r) * N_BANK + col] = acc3[r] * sv;
  }
}

// ---------------- Kernel D: top-8 per query row (ties -> lowest index) ----------------
__global__ void cpr_topk(const float* __restrict__ sims, int* __restrict__ topidx) {
  const int b = blockIdx.x, tid = threadIdx.x;
  const float* row = sims + (size_t)b * N_BANK;
  __shared__ float bv[256];
  __shared__ int   bi[256];
  __shared__ int   picked[TOPK];
  for (int it = 0; it < TOPK; it++) {
    float mv = -INFINITY; int mi = 0x7fffffff;
    for (int n = tid; n < N_BANK; n += 256) {
      bool skip = false;
      for (int j = 0; j < it; j++) skip |= (picked[j] == n);
      if (skip) continue;
      float v = row[n];
      if (v > mv || (v == mv && n < mi)) { mv = v; mi = n; }
    }
    bv[tid] = mv; bi[tid] = mi;
    __syncthreads();
    for (int s = 128; s > 0; s >>= 1) {
      if (tid < s) {
        float v2 = bv[tid + s]; int i2 = bi[tid + s];
        if (v2 > bv[tid] || (v2 == bv[tid] && i2 < bi[tid])) { bv[tid] = v2; bi[tid] = i2; }
      }
      __syncthreads();
    }
    if (tid == 0) { picked[it] = bi[0]; topidx[b * TOPK + it] = bi[0]; }
    __syncthreads();
  }
}

// ---------------- Kernel E: async-gather + (x @ W^T + b) WMMA GEMM + LayerNorm + gate ----
// One block per (b,k): 64x256 tile. Template tile staged into LDS ONCE via the gfx1250
// async-to-LDS path (ASYNCcnt), then all 8 waves feed WMMA A-fragments from LDS
// (conflict-free via 260-float row stride). Same LDS buffer is reused for the h staging
// that feeds the cross-wave LayerNorm.
__global__ void cpr_fuse(const float* __restrict__ templ, const int* __restrict__ topidx,
                         const float* __restrict__ W, const float* __restrict__ bias,
                         const float* __restrict__ gamma, const float* __restrict__ beta,
                         const float* __restrict__ gate_logit, float* __restrict__ out) {
  __shared__ float lds[NT * LDS_STRIDE];   // 66,560 B: template tile, then h
  const int b = blockIdx.x >> 3;
  const int k = blockIdx.x & 7;
  const int tid = threadIdx.x;
  const int wave = tid >> 5, lane = tid & 31;
  const int lr = lane & 15;
  const int khalf = (lane >> 4) << 1;
  const int idx = topidx[b * TOPK + k];
  const float* T = templ + (size_t)idx * NT * C_DIM;
  const int nb = wave * 2;            // first of the wave's two N-tiles

  // ---- Stage T (64x256 = 4096 float4) into LDS, 16 B128 transfers per thread ----
  #pragma unroll
  for (int i = 0; i < 16; i++) {
    const int elem = i * 256 + tid;        // float4 index
    const int row  = elem >> 6;            // 64 float4 per row
    const int colv = elem & 63;
    const int loff = row * LDS_STRIDE + colv * 4;     // floats (16B aligned: 260*4=1040)
    const float* gp = T + row * C_DIM + colv * 4;
#if __has_builtin(__builtin_amdgcn_global_load_async_to_lds_b128)
    __builtin_amdgcn_global_load_async_to_lds_b128(
        (__attribute__((address_space(1))) v4i_gcc*)gp,
        (__attribute__((address_space(3))) v4i_gcc*)&lds[loff], 0, 0);
#else
    *(v4f*)&lds[loff] = *(const v4f*)gp;
#endif
  }
#if __has_builtin(__builtin_amdgcn_global_load_async_to_lds_b128) && \
    __has_builtin(__builtin_amdgcn_s_wait_asynccnt)
  __builtin_amdgcn_s_wait_asynccnt(0);
#endif
  __syncthreads();

  v8f acc[4][2];
  #pragma unroll
  for (int mt = 0; mt < 4; mt++)
    #pragma unroll
    for (int j = 0; j < 2; j++) acc[mt][j] = (v8f){};

  for (int k0 = 0; k0 < C_DIM; k0 += 4) {
    const int kk = k0 + khalf;
    v2f a[4], bf[2];
    #pragma unroll
    for (int mt = 0; mt < 4; mt++) {
      const float* tp = &lds[(mt * 16 + lr) * LDS_STRIDE + kk];   // ds_load_b64, no conflicts
      a[mt].x = tp[0]; a[mt].y = tp[1];
    }
    #pragma unroll
    for (int j = 0; j < 2; j++) {
      // B[k][n] = W[n][k]  (h = x @ W^T); per-wave-private 32 columns, read once per block
      const float* wp = W + ((nb + j) * 16 + lr) * C_DIM + kk;
      bf[j].x = wp[0]; bf[j].y = wp[1];
    }
    #pragma unroll
    for (int mt = 0; mt < 4; mt++)
      #pragma unroll
      for (int j = 0; j < 2; j++)
        acc[mt][j] = __builtin_amdgcn_wmma_f32_16x16x4_f32(false, a[mt], false, bf[j],
                                                           (short)0, acc[mt][j], false, false);
  }
  __syncthreads();   // all waves done reading T before h overwrites the buffer

  const int rbase = (lane >> 4) << 3;
  #pragma unroll
  for (int mt = 0; mt < 4; mt++)
    #pragma unroll
    for (int j = 0; j < 2; j++) {
      const int colc = (nb + j) * 16 + lr;
      #pragma unroll
      for (int r = 0; r < 8; r++)
        lds[(mt * 16 + rbase + r) * LDS_STRIDE + colc] = acc[mt][j][r];
    }
  __syncthreads();

  const float gate = 1.0f / (1.0f + expf(-gate_logit[0]));

  // LayerNorm: wave w handles rows 8w..8w+7, 8 channels per lane (stride-32, conflict-free)
  for (int rr = 0; rr < 8; rr++) {
    const int row = wave * 8 + rr;
    float v[8], sum = 0.0f, sq = 0.0f;
    #pragma unroll
    for (int q = 0; q < 8; q++) {
      const int c = q * 32 + lane;
      const float x = lds[row * LDS_STRIDE + c] + bias[c];
      v[q] = x; sum += x; sq += x * x;
    }
    sum = waveReduceSum(sum);
    sq  = waveReduceSum(sq);
    const float mu   = sum * (1.0f / C_DIM);
    const float var  = sq * (1.0f / C_DIM) - mu * mu;
    const float rstd = rsqrtf(var + 1e-5f);
    const size_t obase = ((size_t)b * (TOPK * NT) + (size_t)k * NT + row) * C_DIM;
    #pragma unroll
    for (int q = 0; q < 8; q++) {
      const int c = q * 32 + lane;
      out[obase + c] = ((v[q] - mu) * rstd * gamma[c] + beta[c]) * gate;
    }
  }
  if (k == 0 && tid == 0)
    out[(size_t)B_Q * (TOPK * NT) * C_DIM + b] = gate;   // second output (B,1)
}

extern "C" void kernel_launch(void* const* d_in, const int* in_sizes, int n_in,
                              void* d_out, int out_size, void* d_ws, size_t ws_size,
                              hipStream_t stream) {
  const float* query = (const float*)d_in[0];
  const float* summ  = (const float*)d_in[1];
  const float* templ = (const float*)d_in[2];
  const float* wproj = (const float*)d_in[3];
  const float* bproj = (const float*)d_in[4];
  const float* gamma = (const float*)d_in[5];
  const float* beta  = (const float*)d_in[6];
  const float* glog  = (const float*)d_in[7];
  float* out = (float*)d_out;

  float* ws   = (float*)d_ws;
  float* qn   = ws;                                   // 64*256
  float* sinv = qn + B_Q * C_DIM;                     // 20000
  float* sims = sinv + N_BANK;                        // 64*20000
  int*   tidx = (int*)(sims + (size_t)B_Q * N_BANK);  // 64*8

  cpr_normalize_query<<<B_Q, C_DIM, 0, stream>>>(query, qn);
  cpr_summary_invnorm<<<(N_BANK + 7) / 8, 256, 0, stream>>>(summ, sinv);
  cpr_sims_wmma<<<N_BANK / 16, 32, 0, stream>>>(qn, summ, sinv, sims);
  cpr_topk<<<B_Q, 256, 0, stream>>>(sims, tidx);
  cpr_fuse<<<B_Q * TOPK, 256, 0, stream>>>(templ, tidx, wproj, bproj, gamma, beta, glog, out);
}